// SlidingWindowLayer_5609227288650
// MI455X (gfx1250) — compile-verified
//
#include <hip/hip_runtime.h>
#include <stdint.h>

// Problem constants (fixed by the reference).
#define WIN   1025
#define DIMC  128
#define DTS   64           // stride DT
#define TAU   8
#define TLEN  (1u << 20)   // T per batch row
#define LOUT  (((TLEN - WIN) / DTS) + 1)   // 16368

// Tiling.
#define LB      256                 // output columns per block
#define NTHREADS 256                // 8 waves (wave32)
#define MFULL   (8 * LB + 120)      // staged input elements per tile = 2168
#define LDSPAD(m) ((m) + ((m) >> 6))              // 1 pad dword per 64 -> conflict-free stride-8 reads
#define LDS_DWORDS (MFULL + (MFULL >> 6) + 8)     // ~2210 dwords (~8.8 KB)

__global__ __launch_bounds__(NTHREADS) void sliding_window_gather_kernel(
    const float* __restrict__ x, float* __restrict__ out) {
  __shared__ float smem[LDS_DWORDS];

  const int tid  = threadIdx.x;
  const int tile = blockIdx.x;      // l-tile index
  const int b    = blockIdx.y;      // batch index
  const int l0   = tile * LB;

  // Base of the input range this tile needs: t = l0*DT + 8*m
  const float* xb = x + (size_t)b * TLEN + (size_t)l0 * DTS;
  // Low 32 bits of a flat shared-memory pointer are the LDS byte offset
  // (aperture lives in the high 32 bits per the CDNA5 flat->LDS mapping).
  const uint32_t lds_base = (uint32_t)(uintptr_t)(&smem[0]);

  // Clip staged loads to the end of the input row.
  int mlim = MFULL;
  {
    long long rem = (long long)TLEN - (long long)l0 * DTS;  // elements available from xb
    if (rem < 0) rem = 0;
    long long mm = (rem + 7) / 8;                           // #m with 8*m < rem
    if (mm < (long long)mlim) mlim = (int)mm;
  }

  // ---- Phase 1: async global -> LDS staging (compacting stride-8 gather) ----
  for (int m = tid; m < mlim; m += NTHREADS) {
    const float* gp = xb + (size_t)m * TAU;
    uint32_t laddr = lds_base + (uint32_t)(LDSPAD(m) * 4);
    asm volatile("global_load_async_to_lds_b32 %0, %1, off"
                 :
                 : "v"(laddr), "v"(gp)
                 : "memory");
  }
  // Our own async loads must land in LDS before we signal the barrier.
  asm volatile("s_wait_asynccnt 0x0" ::: "memory");
  __syncthreads();

  // ---- Phase 2: async LDS -> global, coalesced along l ----
  float* ob = out + (size_t)b * DIMC * LOUT;
  for (int f = tid; f < DIMC * LB; f += NTHREADS) {
    const int c = f >> 8;          // f / LB   (LB == 256)
    const int j = f & (LB - 1);    // f % LB   -> lane-consecutive => 128B coalesced stores
    const int l = l0 + j;
    if (l < (int)LOUT) {
      const int m = 8 * j + c;
      uint32_t laddr = lds_base + (uint32_t)(LDSPAD(m) * 4);
      float* gp = ob + (size_t)c * LOUT + l;
      asm volatile("global_store_async_from_lds_b32 %0, %1, off"
                   :
                   : "v"(gp), "v"(laddr)
                   : "memory");
    }
  }
  // Drain async stores before wave exit.
  asm volatile("s_wait_asynccnt 0x0" ::: "memory");
}

extern "C" void kernel_launch(void* const* d_in, const int* in_sizes, int n_in,
                              void* d_out, int out_size, void* d_ws, size_t ws_size,
                              hipStream_t stream) {
  (void)n_in; (void)out_size; (void)d_ws; (void)ws_size;
  const float* x = (const float*)d_in[0];
  float* out = (float*)d_out;

  const int B = in_sizes[0] / (int)TLEN;                 // 16
  const int ntiles = ((int)LOUT + LB - 1) / LB;          // 64

  dim3 grid(ntiles, B);
  dim3 block(NTHREADS);
  sliding_window_gather_kernel<<<grid, block, 0, stream>>>(x, out);
}